// ResNetG_2542620639561
// MI455X (gfx1250) — compile-verified
//
#include <hip/hip_runtime.h>
#include <math.h>

// GCN forward for MI455X (gfx1250, wave32).
// Bandwidth/atomic-bound problem: dense GEMM done with native fp32 WMMA
// (V_WMMA_F32_16X16X4_F32), edge aggregation with global_atomic_add_f32.

typedef __attribute__((ext_vector_type(2))) float v2f;
typedef __attribute__((ext_vector_type(8))) float v8f;

#define DIN 11   // input feature dim
#define FH  64   // hidden dim

__device__ __forceinline__ void atomAddF(float* p, float v) {
  // relaxed, agent scope -> non-returning global_atomic_add_f32
  __hip_atomic_fetch_add(p, v, __ATOMIC_RELAXED, __HIP_MEMORY_SCOPE_AGENT);
}

// deg starts at 1.0 (self loop); xsum (for x_mean) zeroed
__global__ void k_init(float* __restrict__ deg, float* __restrict__ xsum, int N) {
  int i = blockIdx.x * blockDim.x + threadIdx.x;
  if (i < N) deg[i] = 1.0f;
  if (i < FH) xsum[i] = 0.0f;
}

__global__ void k_deg(const int* __restrict__ src, float* __restrict__ deg, int E) {
  int e = blockIdx.x * blockDim.x + threadIdx.x;
  if (e < E) atomAddF(&deg[src[e]], 1.0f);
}

// hl[N,64] = x[N,11] @ W1[11,64] via V_WMMA_F32_16X16X4_F32.
// One wave per 16-row M-tile, covering all four 16-col N-tiles; K padded 11->12.
// A 16x4 fp32 layout: lanes 0-15 M=0..15 {K+0,K+1}, lanes 16-31 {K+2,K+3}.
__global__ void k_gemm1(const float* __restrict__ x, const float* __restrict__ W1,
                        float* __restrict__ hl, int N) {
  int wave = threadIdx.x >> 5;
  int lane = threadIdx.x & 31;
  int mt   = blockIdx.x * 8 + wave;
  int row0 = mt * 16;
  if (row0 >= N) return;                 // wave-uniform; EXEC stays all-ones for WMMA
  int half = lane >> 4;                  // 0 -> K+0/K+1, 1 -> K+2/K+3
  int l16  = lane & 15;
  v8f c[4] = {{}, {}, {}, {}};
  int  mrow = row0 + l16;
  bool mok  = (mrow < N);
#pragma unroll
  for (int kk = 0; kk < 12; kk += 4) {
    int   ka = kk + half * 2;
    float a0 = (mok && ka < DIN)       ? x[(size_t)mrow * DIN + ka]     : 0.0f;
    float a1 = (mok && (ka + 1) < DIN) ? x[(size_t)mrow * DIN + ka + 1] : 0.0f;
    v2f A = {a0, a1};
#pragma unroll
    for (int nt = 0; nt < 4; ++nt) {
      int   col = nt * 16 + l16;
      float b0 = (ka < DIN)       ? W1[ka * FH + col]       : 0.0f;
      float b1 = ((ka + 1) < DIN) ? W1[(ka + 1) * FH + col] : 0.0f;
      v2f B = {b0, b1};
      c[nt] = __builtin_amdgcn_wmma_f32_16x16x4_f32(
          /*neg_a=*/false, A, /*neg_b=*/false, B,
          /*c_mod=*/(short)0, c[nt], /*reuse_a=*/false, /*reuse_b=*/false);
    }
  }
  // C/D layout: lane = N-col (mod 16), VGPR r holds M=r (lanes 0-15) / M=r+8 (lanes 16-31)
#pragma unroll
  for (int nt = 0; nt < 4; ++nt) {
#pragma unroll
    for (int r = 0; r < 8; ++r) {
      int row = row0 + r + half * 8;
      if (row < N) hl[(size_t)row * FH + nt * 16 + l16] = c[nt][r];
    }
  }
}

__global__ void k_dis(float* __restrict__ dis, int N) {
  int i = blockIdx.x * blockDim.x + threadIdx.x;
  if (i < N) dis[i] = rsqrtf(dis[i]);    // deg >= 1 always (self loop)
}

// agg1[i,:] = dis[i] * hl[i,:]   (self-loop contribution; full overwrite)
__global__ void k_selfinit(const float* __restrict__ hl, const float* __restrict__ dis,
                           float* __restrict__ agg1, int N) {
  int t = blockIdx.x * blockDim.x + threadIdx.x;
  int i = t >> 4, q = t & 15;
  if (i < N) {
    float4 v = ((const float4*)hl)[(size_t)i * 16 + q];
    float  d = dis[i];
    float4 r = {v.x * d, v.y * d, v.z * d, v.w * d};
    ((float4*)agg1)[(size_t)i * 16 + q] = r;
  }
}

// per edge: agg1[src,:] += dis[dst] * hl[dst,:]  (16 threads/edge, float4 each)
__global__ void k_edge1(const int* __restrict__ src, const int* __restrict__ dst,
                        const float* __restrict__ hl, const float* __restrict__ dis,
                        float* __restrict__ agg1, int E) {
  int t = blockIdx.x * blockDim.x + threadIdx.x;
  int e = t >> 4, q = t & 15;
  if (e < E) {
    int    s  = src[e], d = dst[e];
    float  dd = dis[d];
    float4 v  = ((const float4*)hl)[(size_t)d * 16 + q];
    float* b  = agg1 + (size_t)s * FH + q * 4;
    atomAddF(b + 0, v.x * dd);
    atomAddF(b + 1, v.y * dd);
    atomAddF(b + 2, v.z * dd);
    atomAddF(b + 3, v.w * dd);
  }
}

// h = relu(dis*agg1 + b1) (overwrites hl); block-partial sums -> xsum via atomics
__global__ void k_post1(const float* __restrict__ agg1, const float* __restrict__ dis,
                        const float* __restrict__ b1, float* __restrict__ h,
                        float* __restrict__ xsum, int N) {
  __shared__ float sred[256];
  int tid  = threadIdx.x;
  int slot = tid >> 6;         // 4 nodes per 256-thread block iteration
  int f    = tid & 63;
  float acc = 0.0f;
  int ngroups = (N + 3) >> 2;
  for (int g = blockIdx.x; g < ngroups; g += gridDim.x) {
    int node = g * 4 + slot;
    if (node < N) {
      float val = fmaxf(dis[node] * agg1[(size_t)node * FH + f] + b1[f], 0.0f);
      h[(size_t)node * FH + f] = val;
      acc += val;
    }
  }
  sred[tid] = acc;
  __syncthreads();
  if (tid < 64) atomAddF(&xsum[f], sred[f] + sred[f + 64] + sred[f + 128] + sred[f + 192]);
}

// z[i] = concat(h,x)[i] @ W2; store zd=dis*z, init agg2 with self loop
__global__ void k_z(const float* __restrict__ h, const float* __restrict__ x,
                    const float* __restrict__ dis, const float* __restrict__ W2,
                    float* __restrict__ zbuf, float* __restrict__ agg2, int N) {
  int i = blockIdx.x * blockDim.x + threadIdx.x;
  if (i >= N) return;
  const float4* h4 = (const float4*)(h + (size_t)i * FH);
  const float4* w4 = (const float4*)W2;
  float sum = 0.0f;
#pragma unroll
  for (int q = 0; q < 16; ++q) {
    float4 hv = h4[q], wv = w4[q];
    sum += hv.x * wv.x + hv.y * wv.y + hv.z * wv.z + hv.w * wv.w;
  }
#pragma unroll
  for (int j = 0; j < DIN; ++j) sum += x[(size_t)i * DIN + j] * W2[FH + j];
  float zd = dis[i] * sum;
  zbuf[i] = zd;
  agg2[i] = zd;
}

__global__ void k_edge2(const int* __restrict__ src, const int* __restrict__ dst,
                        const float* __restrict__ zbuf, float* __restrict__ agg2, int E) {
  int e = blockIdx.x * blockDim.x + threadIdx.x;
  if (e < E) atomAddF(&agg2[src[e]], zbuf[dst[e]]);
}

// single block: gather logits at ready_idx, softmax over K+1, value head
__global__ void k_final(const int* __restrict__ ready, const float* __restrict__ dis,
                        const float* __restrict__ agg2, const float* __restrict__ xsum,
                        const float* __restrict__ b2, const float* __restrict__ Wd,
                        const float* __restrict__ bd, const float* __restrict__ Wv,
                        const float* __restrict__ bv, float* __restrict__ out,
                        int N, int K) {
  __shared__ float slog[1025];
  __shared__ float sred[1024];
  __shared__ float sxm[FH];
  int tid = threadIdx.x;
  if (tid < FH) sxm[tid] = xsum[tid] / (float)N;
  __syncthreads();
  if (tid == 0) {
    float pn = 0.0f, vv = 0.0f;
    for (int f = 0; f < FH; ++f) { pn += sxm[f] * Wd[f]; vv += sxm[f] * Wv[f]; }
    slog[K] = pn + bd[0];
    out[K + 1] = vv + bv[0];
  }
  float b2v = b2[0];
  for (int t = tid; t < K; t += blockDim.x) {
    int id = ready[t];
    slog[t] = dis[id] * agg2[id] + b2v;
  }
  __syncthreads();
  float lmax = -3.4e38f;
  for (int t = tid; t < K + 1; t += blockDim.x) lmax = fmaxf(lmax, slog[t]);
  sred[tid] = lmax;
  __syncthreads();
  for (int o = 512; o > 0; o >>= 1) {
    if (tid < o) sred[tid] = fmaxf(sred[tid], sred[tid + o]);
    __syncthreads();
  }
  float m = sred[0];
  __syncthreads();
  float lsum = 0.0f;
  for (int t = tid; t < K + 1; t += blockDim.x) {
    float ev = __expf(slog[t] - m);
    slog[t] = ev;
    lsum += ev;
  }
  sred[tid] = lsum;
  __syncthreads();
  for (int o = 512; o > 0; o >>= 1) {
    if (tid < o) sred[tid] += sred[tid + o];
    __syncthreads();
  }
  float S = sred[0];
  for (int t = tid; t < K + 1; t += blockDim.x) out[t] = slog[t] / S;
}

extern "C" void kernel_launch(void* const* d_in, const int* in_sizes, int n_in,
                              void* d_out, int out_size, void* d_ws, size_t ws_size,
                              hipStream_t stream) {
  const float* x    = (const float*)d_in[0];
  const int*   ei   = (const int*)d_in[1];
  const int*   rdy  = (const int*)d_in[2];
  const float* W1   = (const float*)d_in[3];
  const float* b1   = (const float*)d_in[4];
  const float* W2   = (const float*)d_in[5];
  const float* b2   = (const float*)d_in[6];
  const float* Wd   = (const float*)d_in[7];
  const float* bd   = (const float*)d_in[8];
  const float* Wv   = (const float*)d_in[9];
  const float* bv   = (const float*)d_in[10];
  float* out = (float*)d_out;

  int N = in_sizes[0] / DIN;
  int E = in_sizes[1] / 2;
  int K = in_sizes[2];
  const int* src = ei;
  const int* dst = ei + E;

  float* ws = (float*)d_ws;
  size_t nf = (size_t)N * FH;
  float* hl   = ws;            // [N,64] hl, later h (post-relu)
  float* agg1 = ws + nf;       // [N,64]
  float* dis  = ws + 2 * nf;   // [N] deg -> dis (in place)
  float* zbuf = dis + N;       // [N]
  float* agg2 = zbuf + N;      // [N]
  float* xsum = agg2 + N;      // [64]

  dim3 blk(256);
  k_init<<<dim3((N + 255) / 256), blk, 0, stream>>>(dis, xsum, N);
  k_deg<<<dim3((E + 255) / 256), blk, 0, stream>>>(src, dis, E);
  k_gemm1<<<dim3(((N + 15) / 16 + 7) / 8), blk, 0, stream>>>(x, W1, hl, N);
  k_dis<<<dim3((N + 255) / 256), blk, 0, stream>>>(dis, N);
  k_selfinit<<<dim3((N * 16 + 255) / 256), blk, 0, stream>>>(hl, dis, agg1, N);
  {
    long long t1 = (long long)E * 16;
    k_edge1<<<dim3((unsigned)((t1 + 255) / 256)), blk, 0, stream>>>(src, dst, hl, dis, agg1, E);
  }
  k_post1<<<dim3(1024), blk, 0, stream>>>(agg1, dis, b1, hl, xsum, N);
  k_z<<<dim3((N + 255) / 256), blk, 0, stream>>>(hl, x, dis, W2, zbuf, agg2, N);
  k_edge2<<<dim3((E + 255) / 256), blk, 0, stream>>>(src, dst, zbuf, agg2, E);
  k_final<<<dim3(1), dim3(1024), 0, stream>>>(rdy, dis, agg2, xsum, b2, Wd, bd, Wv, bv, out, N, K);
}